// BaseSigKernel_74629351735956
// MI455X (gfx1250) — compile-verified
//
#include <hip/hip_runtime.h>

typedef __attribute__((ext_vector_type(2))) float v2f;
typedef __attribute__((ext_vector_type(8))) float v8f;

#define L    256
#define D    64
#define LM1  255        // rows/cols of dK = (L-1)
#define GRID 510        // PDE increment grid = 2*LM1 (dyadic order 1)
#define NT   512        // threads per block (16 wave32 waves)
#define NW   16

// One block per batch element.
// Phase A (per panel): 16 waves compute a 16x256 panel of dK = dx @ dy^T via
//   V_WMMA_F32_16X16X4_F32 (wave w -> column tile w), store to LDS.
// Phase B: consume 32 PDE rows per panel. Each PDE row is a 510-wide
//   prefix-sum: new[j+1] = 1 + sum_{k<=j}( prev[k+1] + prev[k]*(inc[k]-1) ),
//   done as intra-wave shuffle scan + LDS cross-wave combine.
__global__ __launch_bounds__(NT)
void sigkernel_fused(const float* __restrict__ xs,
                     const float* __restrict__ ys,
                     float* __restrict__ out) {
  __shared__ float panel[16 * 256];   // dK rows r0..r0+15, cols 0..255
  __shared__ float prevu[512];        // current u row, indices 0..510 used
  __shared__ float wsum[NW];

  const int b    = blockIdx.x;
  const int tid  = threadIdx.x;
  const int lane = tid & 31;
  const int wid  = tid >> 5;

  const float* xb = xs + (size_t)b * L * D;
  const float* yb = ys + (size_t)b * L * D;

  // ---- persistent B fragments: wave w owns dK columns 16w..16w+15 ----
  // WMMA 16x16x4 f32 fragment layout: lanes 0-15 carry K={k0,k0+1},
  // lanes 16-31 carry K={k0+2,k0+3}. B[k][n] = dy[col0+n][k].
  const int   halfsel = lane >> 4;
  const int   koff    = halfsel * 2;
  const int   colN    = (wid << 4) + (lane & 15);
  const int   colC    = colN < LM1 ? colN : (LM1 - 1);
  const float bmask   = (colN < LM1) ? 1.0f : 0.0f;

  float bf[32];
#pragma unroll
  for (int t = 0; t < 16; ++t) {
    const int k0 = 4 * t + koff;
    bf[2 * t + 0] = bmask * (yb[(colC + 1) * D + k0]     - yb[colC * D + k0]);
    bf[2 * t + 1] = bmask * (yb[(colC + 1) * D + k0 + 1] - yb[colC * D + k0 + 1]);
  }

  // u[0,:] = 1
  prevu[tid] = 1.0f;
  __syncthreads();

  for (int p = 0; p < 16; ++p) {
    // ---------- compute panel p: dK rows 16p..16p+15 ----------
    const int   rowM  = (p << 4) + (lane & 15);
    const int   rowC  = rowM < LM1 ? rowM : (LM1 - 1);
    const float amask = (rowM < LM1) ? 1.0f : 0.0f;

    v8f acc = {0.f, 0.f, 0.f, 0.f, 0.f, 0.f, 0.f, 0.f};
#pragma unroll
    for (int t = 0; t < 16; ++t) {
      const int k0 = 4 * t + koff;
      v2f a, bb;
      a.x  = amask * (xb[(rowC + 1) * D + k0]     - xb[rowC * D + k0]);
      a.y  = amask * (xb[(rowC + 1) * D + k0 + 1] - xb[rowC * D + k0 + 1]);
      bb.x = bf[2 * t + 0];
      bb.y = bf[2 * t + 1];
      acc = __builtin_amdgcn_wmma_f32_16x16x4_f32(false, a, false, bb,
                                                  (short)0, acc, false, false);
    }

    // D fragment -> LDS. VGPR v: lanes 0-15 => M=v, lanes 16-31 => M=v+8.
    const int nIdx  = (wid << 4) + (lane & 15);
    const int mbase = halfsel * 8;
#pragma unroll
    for (int v = 0; v < 8; ++v)
      panel[(mbase + v) * 256 + nIdx] = acc[v];

    __syncthreads();

    // ---------- PDE rows i = 32p .. 32p+imax-1 ----------
    const int imax = (GRID - 32 * p) < 32 ? (GRID - 32 * p) : 32;
    for (int il = 0; il < imax; ++il) {
      const int rr = il >> 1;                       // dyadic: two PDE rows per dK row
      const float a   = panel[rr * 256 + (tid >> 1)]; // dyadic: col j -> dK col j/2
      const float c   = a * 0.25f - 1.0f;             // inc = dK/4 ; (inc - 1)
      const float pj  = prevu[tid];
      const float pj1 = prevu[tid + 1 < 512 ? tid + 1 : 511];
      float v = (tid < GRID) ? (pj1 + pj * c) : 0.0f;

      // inclusive wave32 scan
#pragma unroll
      for (int dlt = 1; dlt < 32; dlt <<= 1) {
        const float u = __shfl_up(v, dlt, 32);
        if (lane >= dlt) v += u;
      }
      if (lane == 31) wsum[wid] = v;
      __syncthreads();                              // wsum ready; prev reads done

      float off = 0.0f;
      for (int k2 = 0; k2 < wid; ++k2) off += wsum[k2];
      if (tid < GRID) prevu[tid + 1] = 1.0f + off + v;   // prevu[0] stays 1
      __syncthreads();                              // row committed
    }
  }

  if (tid == 0) out[b] = prevu[GRID];
}

extern "C" void kernel_launch(void* const* d_in, const int* in_sizes, int n_in,
                              void* d_out, int out_size, void* d_ws, size_t ws_size,
                              hipStream_t stream) {
  const float* xs = (const float*)d_in[0];
  const float* ys = (const float*)d_in[1];
  float* out = (float*)d_out;
  const int B = in_sizes[0] / (L * D);
  sigkernel_fused<<<B, NT, 0, stream>>>(xs, ys, out);
}